// SatelliteSpecificNormalization_23072564314673
// MI455X (gfx1250) — compile-verified
//
#include <hip/hip_runtime.h>

// Satellite-specific per-(B,N) affine: out = valid ? x*w[sid] + b[sid] : x
// Purely memory-bound (0.25 FLOP/byte). Strategy: b128 streaming, scalar
// uniform sid/w/b, NT stores (preserve 128MiB input residency in 192MB L2),
// async global->LDS pipeline as the CDNA5 bulk-load path.

typedef float v4f __attribute__((ext_vector_type(4)));
typedef int   v4i __attribute__((ext_vector_type(4)));   // builtin's pointee type

#if defined(__HIP_DEVICE_COMPILE__) &&                                         \
    __has_builtin(__builtin_amdgcn_global_load_async_to_lds_b128) &&           \
    __has_builtin(__builtin_amdgcn_s_wait_asynccnt)
#define SAT_USE_ASYNC_LDS 1
#else
#define SAT_USE_ASYNC_LDS 0
#endif

__global__ __launch_bounds__(256)
void SatelliteSpecificNormalization_affine_kernel(
    const float* __restrict__ x, const int* __restrict__ ids,
    const float* __restrict__ weight, const float* __restrict__ bias,
    float* __restrict__ out, int sliceF4) {
  const int bn  = blockIdx.y;                    // (b,n) slice id, 0..127
  const int sid = ids[bn];                       // uniform -> scalar load
  float wv = 1.0f, bv = 0.0f;                    // sid < 0 => identity
  if (sid >= 0) {                                // uniform branch (C == 1)
    wv = weight[sid];
    bv = bias[sid];
  }

  const v4f* __restrict__ xin = (const v4f*)x + (size_t)bn * (size_t)sliceF4;
  v4f* __restrict__ o         = (v4f*)out      + (size_t)bn * (size_t)sliceF4;

  const int tid    = threadIdx.x;
  int       i      = blockIdx.x * blockDim.x + tid;
  const int stride = blockDim.x * gridDim.x;

#if SAT_USE_ASYNC_LDS
  // Depth-2 async pipeline: ASYNCcnt-tracked DMA into per-lane LDS slots.
  __shared__ v4f stage[2][256];                  // 8 KB of the 320 KB WGP LDS
  int buf = 0;
  if (i < sliceF4) {
    __builtin_amdgcn_global_load_async_to_lds_b128(
        (v4i*)(xin + i), (v4i*)&stage[buf][tid], 0, 0);
  }
  for (; i < sliceF4; i += stride) {
    const int nxt = i + stride;
    if (nxt < sliceF4) {
      __builtin_amdgcn_global_load_async_to_lds_b128(
          (v4i*)(xin + nxt), (v4i*)&stage[buf ^ 1][tid], 0, 0);
      __builtin_amdgcn_s_wait_asynccnt(1);       // older request done (in-order)
    } else {
      __builtin_amdgcn_s_wait_asynccnt(0);
    }
    v4f v = stage[buf][tid];                     // ds_load_b128, own slot only
    v4f r = v * wv + bv;                         // contracts to v_fma_f32
    __builtin_nontemporal_store(r, o + i);       // b128 store, th:NT -> keep x in L2
    buf ^= 1;
  }
#else
  // Direct path: b128 load (RT, L2-resident) -> FMA -> b128 NT store.
  for (; i < sliceF4; i += stride) {
    v4f v = xin[i];
    v4f r = v * wv + bv;
    __builtin_nontemporal_store(r, o + i);
  }
#endif
}

extern "C" void kernel_launch(void* const* d_in, const int* in_sizes, int n_in,
                              void* d_out, int out_size, void* d_ws, size_t ws_size,
                              hipStream_t stream) {
  const float* x   = (const float*)d_in[0];   // (16,8,1,512,512) f32
  const int*   ids = (const int*)d_in[1];     // (16,8) i32 in [-1,8)
  const float* w   = (const float*)d_in[2];   // (8,1,1,1) f32
  const float* b   = (const float*)d_in[3];   // (8,1,1,1) f32
  float*       out = (float*)d_out;

  const int BN      = in_sizes[1];            // 128 slices
  const int slice   = in_sizes[0] / BN;       // 262144 floats per slice
  const int sliceF4 = slice / 4;              // 65536 float4 per slice

  const int threads = 256;                    // 8 wave32s per workgroup
  int gridX = sliceF4 / (threads * 4);        // 4 float4 per thread -> 64
  if (gridX < 1) gridX = 1;

  dim3 grid(gridX, BN);                       // 8192 workgroups total
  hipLaunchKernelGGL(SatelliteSpecificNormalization_affine_kernel,
                     grid, dim3(threads), 0, stream,
                     x, ids, w, b, out, sliceF4);
}